// SelfCorrectingPointProcess_23063974380163
// MI455X (gfx1250) — compile-verified
//
#include <hip/hip_runtime.h>
#include <hip/hip_bf16.h>
#include <math.h>

// ---------------------------------------------------------------------------
// SelfCorrectingPointProcess for MI455X (gfx1250, wave32).
//
// Memory-bound: 64MB read + 128MB write => ~8.2us floor at 23.3 TB/s.
// Design: 1 block per row; row staged to LDS via GLOBAL_LOAD_ASYNC_TO_LDS_B128
// (ASYNCcnt path), per-element local reformulation of the telescoping
// compensator sum, wave reductions on the matrix pipe via
// V_WMMA_F32_16X16X4_F32 (B = ones trick), b128 coalesced stores.
// ---------------------------------------------------------------------------

typedef __attribute__((ext_vector_type(2))) float v2f;
typedef __attribute__((ext_vector_type(8))) float v8f;

#define TLEN 4096   // T (fixed by reference)
#define BLK  256    // 8 wave32 per block
#define CH   4      // float4 chunks per thread: 4*256*4 = 4096 elements

// Sum of a value across all 32 lanes of a wave, computed on the matrix pipe.
// Stage 1: A=(p,0), B=ones  -> D[m][*] = p[m] + p[m+16]   (16 partials t[m])
// Stage 2: A=(d[2k],d[2k+1]), B=ones -> t[2k]+t[2k+1]+t[2k+8]+t[2k+9],
//          lane-uniform across the whole wave. 3 adds finish it.
// EXEC must be all-ones here: called with the full block active, pre-divergence.
__device__ __forceinline__ float wave_sum_wmma(float p) {
  v2f ones; ones[0] = 1.0f; ones[1] = 1.0f;
  v8f zc = {};
  v2f a; a[0] = p; a[1] = 0.0f;
  v8f d1 = __builtin_amdgcn_wmma_f32_16x16x4_f32(false, a,   false, ones, (short)0, zc, false, false);
  v2f b01; b01[0] = d1[0]; b01[1] = d1[1];
  v2f b23; b23[0] = d1[2]; b23[1] = d1[3];
  v2f b45; b45[0] = d1[4]; b45[1] = d1[5];
  v2f b67; b67[0] = d1[6]; b67[1] = d1[7];
  v8f r0 = __builtin_amdgcn_wmma_f32_16x16x4_f32(false, b01, false, ones, (short)0, zc, false, false);
  v8f r1 = __builtin_amdgcn_wmma_f32_16x16x4_f32(false, b23, false, ones, (short)0, zc, false, false);
  v8f r2 = __builtin_amdgcn_wmma_f32_16x16x4_f32(false, b45, false, ones, (short)0, zc, false, false);
  v8f r3 = __builtin_amdgcn_wmma_f32_16x16x4_f32(false, b67, false, ones, (short)0, zc, false, false);
  return (r0[0] + r1[0]) + (r2[0] + r3[0]);
}

__global__ __launch_bounds__(BLK) void scpp_kernel(
    const float* __restrict__ times,     // N*T event times
    const float* __restrict__ mu_raw,    // 1
    const float* __restrict__ beta_raw,  // 1
    float* __restrict__ out,             // N + N*T + N*T
    int N)
{
  // Row buffer (16KB) + cross-wave reduction scratch at the tail.
  __shared__ __align__(16) float smem[TLEN + 16];

  const int row = blockIdx.x;
  const int tid = threadIdx.x;

  // Uniform scalar parameters (softplus) and derived constants.
  const float mu    = logf(1.0f + expf(mu_raw[0]));
  const float beta  = logf(1.0f + expf(beta_raw[0]));
  const float L2E   = 1.4426950408889634f;
  const float cE    = mu * L2E;        // exp(mu*t)    = exp2(cE*t)
  const float cW    = -beta * L2E;     // exp(-beta*i) = exp2(cW*i)
  const float Kc    = expf(beta) - 1.0f;
  const float invmu = 1.0f / mu;
  const float wTm2  = __builtin_amdgcn_exp2f(cW * (float)(TLEN - 2));

  const float* trow = times + (size_t)row * TLEN;

  // --- Stage row into LDS on the async path (ASYNCcnt, no VGPR returns). ---
  // smem is the only __shared__ object -> LDS offset 0 of this allocation.
#pragma unroll
  for (int c = 0; c < CH; ++c) {
    const int idx4 = tid + BLK * c;                       // float4 index
    const unsigned lds_off = (unsigned)(idx4 * 16);       // byte offset in LDS
    const unsigned long long ga =
        (unsigned long long)(const void*)(trow + idx4 * 4);
    asm volatile("global_load_async_to_lds_b128 %0, %1, off"
                 :: "v"(lds_off), "v"(ga) : "memory");
  }
  asm volatile("s_wait_asynccnt 0x0" ::: "memory");
  __syncthreads();

  const float4* buf4 = (const float4*)smem;

  // --- Pass 1: per-thread partials for loglik and compensator. ---
  // comp*mu = -E[0] + sum_{1..T-2} (e^b - 1) e^{-b i} E[i] + e^{-b(T-2)} E[T-1]
  float p_ll = 0.0f, p_cmp = 0.0f;
#pragma unroll
  for (int c = 0; c < CH; ++c) {
    const int idx4 = tid + BLK * c;
    const float4 t4 = buf4[idx4];
    const float tv[4] = {t4.x, t4.y, t4.z, t4.w};
#pragma unroll
    for (int e = 0; e < 4; ++e) {
      const int   i  = idx4 * 4 + e;
      const float fi = (float)i;
      const float t  = tv[e];
      p_ll += fmaf(mu, t, -beta * fi);                       // mu*t - beta*i
      const float E = __builtin_amdgcn_exp2f(cE * t);        // exp(mu*t)
      const float w = __builtin_amdgcn_exp2f(cW * fi);       // exp(-beta*i)
      float coef = (i == 0) ? -1.0f : Kc * w;
      if (i == TLEN - 1) coef = wTm2;
      p_cmp = fmaf(coef, E, p_cmp);
    }
  }

  // --- Wave reduction on the matrix pipe (EXEC all-ones, no divergence yet). ---
  const float s_ll  = wave_sum_wmma(p_ll);
  const float s_cmp = wave_sum_wmma(p_cmp);

  // --- Cross-wave reduction (8 waves) through LDS tail. ---
  float* red = smem + TLEN;
  const int lane = tid & 31;
  const int wid  = tid >> 5;
  if (lane == 0) { red[2 * wid] = s_ll; red[2 * wid + 1] = s_cmp; }
  __syncthreads();
  float tot_ll = 0.0f, tot_cmp = 0.0f;
#pragma unroll
  for (int w = 0; w < BLK / 32; ++w) {
    tot_ll  += red[2 * w];
    tot_cmp += red[2 * w + 1];
  }

  const float comp  = tot_cmp * invmu;
  const float scale = __builtin_amdgcn_exp2f(-comp * L2E);   // exp(-comp)

  if (tid == 0) out[row] = tot_ll - comp;                    // output 0: (N,)

  float4* dist4 = (float4*)(out + N) + (size_t)row * (TLEN / 4);
  float4* lamb4 = (float4*)(out + N + (size_t)N * TLEN) + (size_t)row * (TLEN / 4);

  // --- Pass 2: lamb = exp(mu*t - beta*i), dist = lamb * exp(-comp). ---
#pragma unroll
  for (int c = 0; c < CH; ++c) {
    const int idx4 = tid + BLK * c;
    const float4 t4 = buf4[idx4];
    const float tv[4] = {t4.x, t4.y, t4.z, t4.w};
    float lam[4];
#pragma unroll
    for (int e = 0; e < 4; ++e) {
      const float fi = (float)(idx4 * 4 + e);
      lam[e] = __builtin_amdgcn_exp2f(fmaf(cE, tv[e], cW * fi));
    }
    float4 l4, d4;
    l4.x = lam[0]; l4.y = lam[1]; l4.z = lam[2]; l4.w = lam[3];
    d4.x = lam[0] * scale; d4.y = lam[1] * scale;
    d4.z = lam[2] * scale; d4.w = lam[3] * scale;
    lamb4[idx4] = l4;
    dist4[idx4] = d4;
  }
}

extern "C" void kernel_launch(void* const* d_in, const int* in_sizes, int n_in,
                              void* d_out, int out_size, void* d_ws, size_t ws_size,
                              hipStream_t stream) {
  // Inputs (setup_inputs order): input_time, input_loc, input_mag,
  // input_timediff, mu_raw, beta_raw. Only time + the two scalars are used.
  const float* times    = (const float*)d_in[0];
  const float* mu_raw   = (const float*)d_in[4];
  const float* beta_raw = (const float*)d_in[5];
  float* out = (float*)d_out;
  const int N = in_sizes[0] / TLEN;
  scpp_kernel<<<N, BLK, 0, stream>>>(times, mu_raw, beta_raw, out, N);
}